// SEGNNLayer_64793876627490
// MI455X (gfx1250) — compile-verified
//
#include <hip/hip_runtime.h>
#include <math.h>

// ---------------------------------------------------------------------------
// SEGNN layer for gfx1250 (MI455X), fp32 WMMA (V_WMMA_F32_16X16X4_F32).
//
// Input order (setup_inputs):
//  0 node_s [N,64]        1 node_v [N,64,3]    2 node_attr0 [N]   3 node_attr1 [N,3]
//  4 edge_attr0 [E]       5 edge_attr1 [E,3]   6 extra_feats [E,2]
//  7 w_m0_0 [258,128]     8 w_m0_1 [258,64]    9 b_m0 [128]
// 10 w_m1_0 [128,128]    11 w_m1_1 [128,64]   12 b_m1 [128]
// 13 w_u0_0 [256,128]    14 w_u0_1 [256,64]   15 b_u0 [128]
// 16 w_u1_0 [128,64]     17 w_u1_1 [128,64]   18 b_u1 [64]
// 19 senders [E] i32     20 receivers [E] i32
// Output: [N, 256] = concat(new_s [N,64], new_v [N,64,3] flattened)
// ---------------------------------------------------------------------------

typedef float v2f __attribute__((ext_vector_type(2)));
typedef float v8f __attribute__((ext_vector_type(8)));

#define INV3 0.57735026918962576f   // 1/sqrt(3)

// path normalization constants: c0 = 1/sqrt(fan), c1 = sqrt(3/fan)
#define C0_F258 0.06225728063f
#define C1_F258 0.10783277320f
#define C0_F256 0.0625f
#define C1_F256 0.10825317547f
#define C0_F128 0.08838834765f
#define C1_F128 0.15309310892f

__device__ __forceinline__ v8f wmma4(v2f a, v2f b, v8f c) {
  // D = A(16x4 f32) * B(4x16 f32) + C(16x16 f32), wave32
  return __builtin_amdgcn_wmma_f32_16x16x4_f32(false, a, false, b, (short)0, c,
                                               false, false);
}

__device__ __forceinline__ float sigm(float x) { return 1.0f / (1.0f + __expf(-x)); }
__device__ __forceinline__ float silu(float x) { return x * sigm(x); }

// z-row pair for one k-chunk: a[] for k0, b[] for k0+1; index 0 = z0 (scalar
// output path), 1..3 = z1 x/y/z (vector output path).
struct Z4 {
  float a[4];
  float b[4];
};

__device__ __forceinline__ Z4 z_scalar(float x0, float x1, float a0v,
                                       float axi, float ayi, float azi) {
  Z4 z;
  z.a[0] = x0 * a0v; z.a[1] = x0 * axi; z.a[2] = x0 * ayi; z.a[3] = x0 * azi;
  z.b[0] = x1 * a0v; z.b[1] = x1 * axi; z.b[2] = x1 * ayi; z.b[3] = x1 * azi;
  return z;
}

__device__ __forceinline__ Z4 z_vector(const float* __restrict__ p,  // 6 floats
                                       float a0i, float axi, float ayi,
                                       float azi) {
  Z4 z;
  float v0x = p[0], v0y = p[1], v0z = p[2];
  float v1x = p[3], v1y = p[4], v1z = p[5];
  z.a[0] = v0x * axi + v0y * ayi + v0z * azi;
  z.a[1] = v0x * a0i; z.a[2] = v0y * a0i; z.a[3] = v0z * a0i;
  z.b[0] = v1x * axi + v1y * ayi + v1z * azi;
  z.b[1] = v1x * a0i; z.b[2] = v1y * a0i; z.b[3] = v1z * a0i;
  return z;
}

// 5-WMMA step: two 16-col tiles of the scalar output GEMM (W0, ld0 columns)
// plus one 16-col tile x 3 components of the vector output GEMM (W1, 64 cols).
__device__ __forceinline__ void step5(int k0, const Z4& z,
                                      const float* __restrict__ W0, int ld0,
                                      int col0a, int col0b,
                                      const float* __restrict__ W1, int col1,
                                      v8f acc0[2], v8f acc1[3]) {
  v2f a0; a0.x = z.a[0]; a0.y = z.b[0];
  {
    v2f b; b.x = W0[k0 * ld0 + col0a]; b.y = W0[(k0 + 1) * ld0 + col0a];
    acc0[0] = wmma4(a0, b, acc0[0]);
  }
  {
    v2f b; b.x = W0[k0 * ld0 + col0b]; b.y = W0[(k0 + 1) * ld0 + col0b];
    acc0[1] = wmma4(a0, b, acc0[1]);
  }
  v2f b1; b1.x = W1[k0 * 64 + col1]; b1.y = W1[(k0 + 1) * 64 + col1];
#pragma unroll
  for (int c = 0; c < 3; ++c) {
    v2f a; a.x = z.a[1 + c]; a.y = z.b[1 + c];
    acc1[c] = wmma4(a, b1, acc1[c]);
  }
}

// 4-WMMA step: one 16-col tile of scalar GEMM + one tile x 3 comps of vector.
__device__ __forceinline__ void step4(int k0, const Z4& z,
                                      const float* __restrict__ W0,
                                      const float* __restrict__ W1, int col,
                                      v8f& acc0, v8f acc1[3]) {
  {
    v2f a; a.x = z.a[0]; a.y = z.b[0];
    v2f b; b.x = W0[k0 * 64 + col]; b.y = W0[(k0 + 1) * 64 + col];
    acc0 = wmma4(a, b, acc0);
  }
  v2f b1; b1.x = W1[k0 * 64 + col]; b1.y = W1[(k0 + 1) * 64 + col];
#pragma unroll
  for (int c = 0; c < 3; ++c) {
    v2f a; a.x = z.a[1 + c]; a.y = z.b[1 + c];
    acc1[c] = wmma4(a, b1, acc1[c]);
  }
}

// ---------------------------------------------------------------------------
__global__ void segnn_zero_kernel(float* __restrict__ p, int n) {
  int i = blockIdx.x * blockDim.x + threadIdx.x;
  int stride = gridDim.x * blockDim.x;
  for (; i < n; i += stride) p[i] = 0.0f;
}

// ---------------------------------------------------------------------------
// Edge kernel: one 16-edge tile per 128-thread block (4 wave32s).
// Block-1 TP (fan=258) + gate, block-2 TP (fan=128) + gate, atomic scatter-add.
// ---------------------------------------------------------------------------
__global__ __launch_bounds__(128) void segnn_edge_kernel(
    const float* __restrict__ node_s, const float* __restrict__ node_v,
    const float* __restrict__ ea0g, const float* __restrict__ ea1g,
    const float* __restrict__ exg,
    const float* __restrict__ W00, const float* __restrict__ W01,
    const float* __restrict__ B0,
    const float* __restrict__ W10, const float* __restrict__ W11,
    const float* __restrict__ B1,
    const int* __restrict__ senders, const int* __restrict__ receivers,
    float* __restrict__ agg_s, float* __restrict__ agg_v, int E) {
  __shared__ float ss[16 * 64];        // sender scalars
  __shared__ float sv[16 * 64 * 3];    // sender vectors
  __shared__ float rs[16 * 64];        // receiver scalars
  __shared__ float rv[16 * 64 * 3];    // receiver vectors
  __shared__ float exl[16 * 2];
  __shared__ float ea0[16];
  __shared__ float ea1[16 * 3];
  __shared__ int send_l[16];
  __shared__ int recv_l[16];           // -1 => padded (invalid) edge
  __shared__ float s1[16 * 128];       // scalar TP output (pre-gate)
  __shared__ float sg[16 * 64];        // gated scalars
  __shared__ float vg[3 * 16 * 64];    // gated vectors [comp][row][m]

  const int tid = threadIdx.x;
  const int ebase = blockIdx.x * 16;

  if (tid < 16) {
    int e = ebase + tid;
    bool valid = (e < E);
    send_l[tid] = valid ? senders[e] : 0;
    recv_l[tid] = valid ? receivers[e] : -1;
    ea0[tid] = valid ? ea0g[e] : 0.0f;   // zero attrs neutralize padded rows
    ea1[tid * 3 + 0] = valid ? ea1g[e * 3 + 0] : 0.0f;
    ea1[tid * 3 + 1] = valid ? ea1g[e * 3 + 1] : 0.0f;
    ea1[tid * 3 + 2] = valid ? ea1g[e * 3 + 2] : 0.0f;
    exl[tid * 2 + 0] = valid ? exg[e * 2 + 0] : 0.0f;
    exl[tid * 2 + 1] = valid ? exg[e * 2 + 1] : 0.0f;
  }
  __syncthreads();

  for (int i = tid; i < 16 * 64; i += 128) {
    int row = i >> 6, f = i & 63;
    int sidx = send_l[row];
    int ridx = recv_l[row] < 0 ? 0 : recv_l[row];
    ss[i] = node_s[sidx * 64 + f];
    rs[i] = node_s[ridx * 64 + f];
  }
  for (int i = tid; i < 16 * 192; i += 128) {
    int row = i / 192, f = i % 192;
    int sidx = send_l[row];
    int ridx = recv_l[row] < 0 ? 0 : recv_l[row];
    sv[i] = node_v[sidx * 192 + f];
    rv[i] = node_v[ridx * 192 + f];
  }
  __syncthreads();

  const int lane = tid & 31;
  const int wave = tid >> 5;
  const int arow = lane & 15;          // A row / B col within tile
  const int khalf = (lane >> 4) << 1;  // 0 or 2
  const int rbase = (lane >> 4) << 3;  // C/D row base
  const int cd = lane & 15;            // C/D col within tile
  const int col0a = wave * 32 + arow;  // scalar-output cols (two tiles)
  const int col0b = col0a + 16;
  const int col1 = wave * 16 + arow;   // vector-output col

  // loop-invariant attributes for this lane's A-row
  const float a0v = ea0[arow];
  const float axi = ea1[arow * 3 + 0] * INV3;
  const float ayi = ea1[arow * 3 + 1] * INV3;
  const float azi = ea1[arow * 3 + 2] * INV3;
  const float a0i = a0v * INV3;

  // ---------------- block 1: fan = 258 -------------------------------------
  v8f acc0[2] = {};
  v8f acc1[3] = {};
  // seg 0: sender scalars, k in [0,64)
  for (int chunk = 0; chunk < 16; ++chunk) {
    int k0 = chunk * 4 + khalf;
    Z4 z = z_scalar(ss[arow * 64 + k0], ss[arow * 64 + k0 + 1], a0v, axi, ayi, azi);
    step5(k0, z, W00, 128, col0a, col0b, W01, col1, acc0, acc1);
  }
  // seg 1: sender vectors, k in [64,128)
  for (int chunk = 16; chunk < 32; ++chunk) {
    int k0 = chunk * 4 + khalf;
    Z4 z = z_vector(&sv[(arow * 64 + (k0 - 64)) * 3], a0i, axi, ayi, azi);
    step5(k0, z, W00, 128, col0a, col0b, W01, col1, acc0, acc1);
  }
  // seg 2: receiver scalars, k in [128,192)
  for (int chunk = 32; chunk < 48; ++chunk) {
    int k0 = chunk * 4 + khalf;
    int ks = k0 - 128;
    Z4 z = z_scalar(rs[arow * 64 + ks], rs[arow * 64 + ks + 1], a0v, axi, ayi, azi);
    step5(k0, z, W00, 128, col0a, col0b, W01, col1, acc0, acc1);
  }
  // seg 3: receiver vectors, k in [192,256)
  for (int chunk = 48; chunk < 64; ++chunk) {
    int k0 = chunk * 4 + khalf;
    Z4 z = z_vector(&rv[(arow * 64 + (k0 - 192)) * 3], a0i, axi, ayi, azi);
    step5(k0, z, W00, 128, col0a, col0b, W01, col1, acc0, acc1);
  }
  // tail: extra feats, k in [256,258); lanes with khalf==2 map to pad -> 0
  {
    const bool lo = (khalf == 0);
    int k0 = 256 + khalf;
    int k0c = lo ? k0 : 256;       // clamp so loads stay in bounds
    float x0 = lo ? exl[arow * 2 + 0] : 0.0f;
    float x1 = lo ? exl[arow * 2 + 1] : 0.0f;
    Z4 z = z_scalar(x0, x1, a0v, axi, ayi, azi);
    step5(k0c, z, W00, 128, col0a, col0b, W01, col1, acc0, acc1);
  }

  // scalar output -> LDS (scaled + bias)
#pragma unroll
  for (int t = 0; t < 2; ++t) {
    int col = wave * 32 + t * 16 + cd;
#pragma unroll
    for (int r = 0; r < 8; ++r)
      s1[(rbase + r) * 128 + col] = acc0[t][r] * C0_F258 + B0[col];
  }
  __syncthreads();

  // gate: features = silu(s1[:, :64]); vectors *= sigmoid(s1[:, 64:])
  {
    int vcol = wave * 16 + cd;
#pragma unroll
    for (int r = 0; r < 8; ++r) {
      int rd = rbase + r;
      float g = sigm(s1[rd * 128 + 64 + vcol]);
#pragma unroll
      for (int c = 0; c < 3; ++c)
        vg[c * 1024 + rd * 64 + vcol] = acc1[c][r] * C1_F258 * g;
    }
  }
  if (wave < 2) {
#pragma unroll
    for (int t = 0; t < 2; ++t) {
      int col = wave * 32 + t * 16 + cd;
#pragma unroll
      for (int r = 0; r < 8; ++r) {
        int rd = rbase + r;
        sg[rd * 64 + col] = silu(s1[rd * 128 + col]);
      }
    }
  }
  __syncthreads();

  // ---------------- block 2: fan = 128 -------------------------------------
  v8f bcc0[2] = {};
  v8f bcc1[3] = {};
  // scalar segment, k in [0,64)
  for (int chunk = 0; chunk < 16; ++chunk) {
    int k0 = chunk * 4 + khalf;
    Z4 z = z_scalar(sg[arow * 64 + k0], sg[arow * 64 + k0 + 1], a0v, axi, ayi, azi);
    step5(k0, z, W10, 128, col0a, col0b, W11, col1, bcc0, bcc1);
  }
  // vector segment, k in [64,128); vg is [comp][row][m]
  for (int chunk = 16; chunk < 32; ++chunk) {
    int k0 = chunk * 4 + khalf;
    int m = k0 - 64;
    float vx0 = vg[arow * 64 + m],        vx1 = vg[arow * 64 + m + 1];
    float vy0 = vg[1024 + arow * 64 + m], vy1 = vg[1024 + arow * 64 + m + 1];
    float vz0 = vg[2048 + arow * 64 + m], vz1 = vg[2048 + arow * 64 + m + 1];
    Z4 z;
    z.a[0] = vx0 * axi + vy0 * ayi + vz0 * azi;
    z.a[1] = vx0 * a0i; z.a[2] = vy0 * a0i; z.a[3] = vz0 * a0i;
    z.b[0] = vx1 * axi + vy1 * ayi + vz1 * azi;
    z.b[1] = vx1 * a0i; z.b[2] = vy1 * a0i; z.b[3] = vz1 * a0i;
    step5(k0, z, W10, 128, col0a, col0b, W11, col1, bcc0, bcc1);
  }

#pragma unroll
  for (int t = 0; t < 2; ++t) {
    int col = wave * 32 + t * 16 + cd;
#pragma unroll
    for (int r = 0; r < 8; ++r)
      s1[(rbase + r) * 128 + col] = bcc0[t][r] * C0_F128 + B1[col];
  }
  __syncthreads();

  // gate + atomic scatter-add into receiver aggregates
  {
    int vcol = wave * 16 + cd;
#pragma unroll
    for (int r = 0; r < 8; ++r) {
      int rd = rbase + r;
      int rcv = recv_l[rd];
      if (rcv >= 0) {
        float g = sigm(s1[rd * 128 + 64 + vcol]);
#pragma unroll
        for (int c = 0; c < 3; ++c)
          __hip_atomic_fetch_add(&agg_v[(rcv * 64 + vcol) * 3 + c],
                                 bcc1[c][r] * C1_F128 * g, __ATOMIC_RELAXED,
                                 __HIP_MEMORY_SCOPE_AGENT);
      }
    }
  }
  if (wave < 2) {
#pragma unroll
    for (int t = 0; t < 2; ++t) {
      int col = wave * 32 + t * 16 + cd;
#pragma unroll
      for (int r = 0; r < 8; ++r) {
        int rd = rbase + r;
        int rcv = recv_l[rd];
        if (rcv >= 0)
          __hip_atomic_fetch_add(&agg_s[rcv * 64 + col],
                                 silu(s1[rd * 128 + col]), __ATOMIC_RELAXED,
                                 __HIP_MEMORY_SCOPE_AGENT);
      }
    }
  }
}

// ---------------------------------------------------------------------------
// Node kernel: one 16-node tile per 128-thread block.
// Gated TP u0 (fan=256), plain TP u1 (fan=128), residual, write output.
// ---------------------------------------------------------------------------
__global__ __launch_bounds__(128) void segnn_node_kernel(
    const float* __restrict__ node_s, const float* __restrict__ node_v,
    const float* __restrict__ na0g, const float* __restrict__ na1g,
    const float* __restrict__ agg_s, const float* __restrict__ agg_v,
    const float* __restrict__ Wu00, const float* __restrict__ Wu01,
    const float* __restrict__ Bu0,
    const float* __restrict__ Wu10, const float* __restrict__ Wu11,
    const float* __restrict__ Bu1, float* __restrict__ out, int N) {
  __shared__ float ns[16 * 64];
  __shared__ float nv[16 * 64 * 3];
  __shared__ float as_[16 * 64];
  __shared__ float av[16 * 64 * 3];
  __shared__ float na0[16];
  __shared__ float na1[16 * 3];
  __shared__ float s1[16 * 128];
  __shared__ float sg[16 * 64];
  __shared__ float vg[3 * 16 * 64];

  const int tid = threadIdx.x;
  const int nbase = blockIdx.x * 16;

  if (tid < 16) {
    int n = nbase + tid;
    bool valid = (n < N);
    na0[tid] = valid ? na0g[n] : 0.0f;   // zero attrs neutralize padded rows
    na1[tid * 3 + 0] = valid ? na1g[n * 3 + 0] : 0.0f;
    na1[tid * 3 + 1] = valid ? na1g[n * 3 + 1] : 0.0f;
    na1[tid * 3 + 2] = valid ? na1g[n * 3 + 2] : 0.0f;
  }
  for (int i = tid; i < 16 * 64; i += 128) {
    int row = i >> 6, f = i & 63;
    int n = nbase + row;
    int nc = n < N ? n : 0;
    ns[i] = node_s[nc * 64 + f];
    as_[i] = agg_s[nc * 64 + f];
  }
  for (int i = tid; i < 16 * 192; i += 128) {
    int row = i / 192, f = i % 192;
    int n = nbase + row;
    int nc = n < N ? n : 0;
    nv[i] = node_v[nc * 192 + f];
    av[i] = agg_v[nc * 192 + f];
  }
  __syncthreads();

  const int lane = tid & 31;
  const int wave = tid >> 5;
  const int arow = lane & 15;
  const int khalf = (lane >> 4) << 1;
  const int rbase = (lane >> 4) << 3;
  const int cd = lane & 15;
  const int col0a = wave * 32 + arow;
  const int col0b = col0a + 16;
  const int col1 = wave * 16 + arow;

  const float a0v = na0[arow];
  const float axi = na1[arow * 3 + 0] * INV3;
  const float ayi = na1[arow * 3 + 1] * INV3;
  const float azi = na1[arow * 3 + 2] * INV3;
  const float a0i = a0v * INV3;

  // ---------------- block u0: fan = 256 ------------------------------------
  v8f acc0[2] = {};
  v8f acc1[3] = {};
  for (int chunk = 0; chunk < 16; ++chunk) {  // node scalars
    int k0 = chunk * 4 + khalf;
    Z4 z = z_scalar(ns[arow * 64 + k0], ns[arow * 64 + k0 + 1], a0v, axi, ayi, azi);
    step5(k0, z, Wu00, 128, col0a, col0b, Wu01, col1, acc0, acc1);
  }
  for (int chunk = 16; chunk < 32; ++chunk) {  // node vectors
    int k0 = chunk * 4 + khalf;
    Z4 z = z_vector(&nv[(arow * 64 + (k0 - 64)) * 3], a0i, axi, ayi, azi);
    step5(k0, z, Wu00, 128, col0a, col0b, Wu01, col1, acc0, acc1);
  }
  for (int chunk = 32; chunk < 48; ++chunk) {  // aggregated scalars
    int k0 = chunk * 4 + khalf;
    int ks = k0 - 128;
    Z4 z = z_scalar(as_[arow * 64 + ks], as_[arow * 64 + ks + 1], a0v, axi, ayi, azi);
    step5(k0, z, Wu00, 128, col0a, col0b, Wu01, col1, acc0, acc1);
  }
  for (int chunk = 48; chunk < 64; ++chunk) {  // aggregated vectors
    int k0 = chunk * 4 + khalf;
    Z4 z = z_vector(&av[(arow * 64 + (k0 - 192)) * 3], a0i, axi, ayi, azi);
    step5(k0, z, Wu00, 128, col0a, col0b, Wu01, col1, acc0, acc1);
  }

#pragma unroll
  for (int t = 0; t < 2; ++t) {
    int col = wave * 32 + t * 16 + cd;
#pragma unroll
    for (int r = 0; r < 8; ++r)
      s1[(rbase + r) * 128 + col] = acc0[t][r] * C0_F256 + Bu0[col];
  }
  __syncthreads();

  {
    int vcol = wave * 16 + cd;
#pragma unroll
    for (int r = 0; r < 8; ++r) {
      int rd = rbase + r;
      float g = sigm(s1[rd * 128 + 64 + vcol]);
#pragma unroll
      for (int c = 0; c < 3; ++c)
        vg[c * 1024 + rd * 64 + vcol] = acc1[c][r] * C1_F256 * g;
    }
  }
  if (wave < 2) {
#pragma unroll
    for (int t = 0; t < 2; ++t) {
      int col = wave * 32 + t * 16 + cd;
#pragma unroll
      for (int r = 0; r < 8; ++r) {
        int rd = rbase + r;
        sg[rd * 64 + col] = silu(s1[rd * 128 + col]);
      }
    }
  }
  __syncthreads();

  // ---------------- block u1: fan = 128, no gate, residual -----------------
  v8f ucc0 = {};
  v8f ucc1[3] = {};
  for (int chunk = 0; chunk < 16; ++chunk) {  // gated scalars
    int k0 = chunk * 4 + khalf;
    Z4 z = z_scalar(sg[arow * 64 + k0], sg[arow * 64 + k0 + 1], a0v, axi, ayi, azi);
    step4(k0, z, Wu10, Wu11, col1, ucc0, ucc1);
  }
  for (int chunk = 16; chunk < 32; ++chunk) {  // gated vectors
    int k0 = chunk * 4 + khalf;
    int m = k0 - 64;
    float vx0 = vg[arow * 64 + m],        vx1 = vg[arow * 64 + m + 1];
    float vy0 = vg[1024 + arow * 64 + m], vy1 = vg[1024 + arow * 64 + m + 1];
    float vz0 = vg[2048 + arow * 64 + m], vz1 = vg[2048 + arow * 64 + m + 1];
    Z4 z;
    z.a[0] = vx0 * axi + vy0 * ayi + vz0 * azi;
    z.a[1] = vx0 * a0i; z.a[2] = vy0 * a0i; z.a[3] = vz0 * a0i;
    z.b[0] = vx1 * axi + vy1 * ayi + vz1 * azi;
    z.b[1] = vx1 * a0i; z.b[2] = vy1 * a0i; z.b[3] = vz1 * a0i;
    step4(k0, z, Wu10, Wu11, col1, ucc0, ucc1);
  }

  {
    int col = wave * 16 + cd;  // both scalar col and vector m index
#pragma unroll
    for (int r = 0; r < 8; ++r) {
      int rd = rbase + r;
      int n = nbase + rd;
      if (n < N) {
        float us = ucc0[r] * C0_F128 + Bu1[col];
        out[n * 256 + col] = node_s[n * 64 + col] + us;
#pragma unroll
        for (int c = 0; c < 3; ++c)
          out[n * 256 + 64 + col * 3 + c] =
              node_v[(n * 64 + col) * 3 + c] + ucc1[c][r] * C1_F128;
      }
    }
  }
}

// ---------------------------------------------------------------------------
extern "C" void kernel_launch(void* const* d_in, const int* in_sizes, int n_in,
                              void* d_out, int out_size, void* d_ws,
                              size_t ws_size, hipStream_t stream) {
  const float* node_s = (const float*)d_in[0];
  const float* node_v = (const float*)d_in[1];
  const float* node_attr0 = (const float*)d_in[2];
  const float* node_attr1 = (const float*)d_in[3];
  const float* edge_attr0 = (const float*)d_in[4];
  const float* edge_attr1 = (const float*)d_in[5];
  const float* extra_feats = (const float*)d_in[6];
  const float* w_m0_0 = (const float*)d_in[7];
  const float* w_m0_1 = (const float*)d_in[8];
  const float* b_m0 = (const float*)d_in[9];
  const float* w_m1_0 = (const float*)d_in[10];
  const float* w_m1_1 = (const float*)d_in[11];
  const float* b_m1 = (const float*)d_in[12];
  const float* w_u0_0 = (const float*)d_in[13];
  const float* w_u0_1 = (const float*)d_in[14];
  const float* b_u0 = (const float*)d_in[15];
  const float* w_u1_0 = (const float*)d_in[16];
  const float* w_u1_1 = (const float*)d_in[17];
  const float* b_u1 = (const float*)d_in[18];
  const int* senders = (const int*)d_in[19];
  const int* receivers = (const int*)d_in[20];

  const int N = in_sizes[2];  // node_attr0 has shape (N,)
  const int E = in_sizes[4];  // edge_attr0 has shape (E,)

  float* agg_s = (float*)d_ws;               // [N,64]
  float* agg_v = agg_s + (size_t)N * 64;     // [N,64,3]

  segnn_zero_kernel<<<512, 256, 0, stream>>>(agg_s, N * 256);

  segnn_edge_kernel<<<(E + 15) / 16, 128, 0, stream>>>(
      node_s, node_v, edge_attr0, edge_attr1, extra_feats, w_m0_0, w_m0_1,
      b_m0, w_m1_0, w_m1_1, b_m1, senders, receivers, agg_s, agg_v, E);

  segnn_node_kernel<<<(N + 15) / 16, 128, 0, stream>>>(
      node_s, node_v, node_attr0, node_attr1, agg_s, agg_v, w_u0_0, w_u0_1,
      b_u0, w_u1_0, w_u1_1, b_u1, (float*)d_out, N);
}